// GNNEncoder_43671227465977
// MI455X (gfx1250) — compile-verified
//
#include <hip/hip_runtime.h>
#include <hip/hip_bf16.h>
#include <math.h>

typedef __attribute__((ext_vector_type(2))) float v2f;
typedef __attribute__((ext_vector_type(8))) float v8f;

#define DFEAT 64

// ---------------- degree / dinv ----------------
__global__ __launch_bounds__(256) void k_init_deg(float* __restrict__ deg, int n) {
    int i = blockIdx.x * 256 + threadIdx.x;
    if (i < n) deg[i] = 1.0f;   // self-loop weight
}

__global__ __launch_bounds__(256) void k_edge_deg(const int* __restrict__ dst,
                                                  const float* __restrict__ w,
                                                  float* __restrict__ deg, int e) {
    int i = blockIdx.x * 256 + threadIdx.x;
    if (i < e) atomicAdd(&deg[dst[i]], w[i]);
}

__global__ __launch_bounds__(256) void k_dinv(float* __restrict__ deg, int n) {
    int i = blockIdx.x * 256 + threadIdx.x;
    if (i < n) {
        float d = deg[i];
        deg[i] = (d > 0.0f) ? rsqrtf(d) : 0.0f;
    }
}

// ---------------- dense GEMM: H[n,64] = X[n,64] * W[64,64] (WMMA f32 16x16x4) ----
__global__ __launch_bounds__(128) void k_gemm64(const float* __restrict__ X,
                                                const float* __restrict__ W,
                                                float* __restrict__ H, int n) {
    const int lane = threadIdx.x & 31;
    const int wave = threadIdx.x >> 5;
    const int row0 = (blockIdx.x * 4 + wave) * 16;
    if (row0 >= n) return;                 // wave-uniform: EXEC all-ones inside

    const int m  = lane & 15;              // row-in-tile (A) / col-in-tile (B,D)
    const int kh = lane >> 4;              // K-half select
    const float* xrow = X + (size_t)(row0 + m) * DFEAT;

    v8f c0 = {}, c1 = {}, c2 = {}, c3 = {};
    #pragma unroll
    for (int k = 0; k < DFEAT; k += 4) {
        const int kk = k + kh * 2;
        // A fragment: X[row, kk], X[row, kk+1]  (kk even -> 8B aligned)
        const v2f a = *(const v2f*)(xrow + kk);
        // B fragments for 4 column tiles: W[kk, col], W[kk+1, col]
        const float* w0 = W + (size_t)kk * DFEAT;
        const float* w1 = w0 + DFEAT;
        v2f b0, b1, b2, b3;
        b0.x = w0[m +  0]; b0.y = w1[m +  0];
        b1.x = w0[m + 16]; b1.y = w1[m + 16];
        b2.x = w0[m + 32]; b2.y = w1[m + 32];
        b3.x = w0[m + 48]; b3.y = w1[m + 48];
        c0 = __builtin_amdgcn_wmma_f32_16x16x4_f32(false, a, false, b0, (short)0, c0, false, false);
        c1 = __builtin_amdgcn_wmma_f32_16x16x4_f32(false, a, false, b1, (short)0, c1, false, false);
        c2 = __builtin_amdgcn_wmma_f32_16x16x4_f32(false, a, false, b2, (short)0, c2, false, false);
        c3 = __builtin_amdgcn_wmma_f32_16x16x4_f32(false, a, false, b3, (short)0, c3, false, false);
    }
    // D layout: VGPR r -> row r + 8*kh, col = m
    float* hb = H + (size_t)row0 * DFEAT;
    #pragma unroll
    for (int r = 0; r < 8; ++r) {
        const int rr = r + kh * 8;
        float* o = hb + (size_t)rr * DFEAT + m;
        o[ 0] = c0[r];
        o[16] = c1[r];
        o[32] = c2[r];
        o[48] = c3[r];
    }
}

// ---------------- self-loop init: out[i,:] = h[i,:]*dinv[i]^2 (+ bias) --------
__global__ __launch_bounds__(256) void k_selfloop(const float* __restrict__ h,
                                                  const float* __restrict__ dinv,
                                                  const float* __restrict__ bias, // may be null
                                                  float* __restrict__ out, int n) {
    unsigned gid = blockIdx.x * 256u + threadIdx.x;
    int i  = (int)(gid >> 4);
    if (i >= n) return;
    int f4 = (int)(gid & 15u) * 4;
    float di = dinv[i];
    float nrm = di * di;
    const float4 hv = *(const float4*)(h + (size_t)i * DFEAT + f4);
    float4 o;
    o.x = hv.x * nrm; o.y = hv.y * nrm; o.z = hv.z * nrm; o.w = hv.w * nrm;
    if (bias) { o.x += bias[f4]; o.y += bias[f4+1]; o.z += bias[f4+2]; o.w += bias[f4+3]; }
    *(float4*)(out + (size_t)i * DFEAT + f4) = o;
}

// ---------------- edge scatter: out[dst] += h[src]*dinv[src]*w*dinv[dst] ------
__global__ __launch_bounds__(256) void k_edge_scatter(const float* __restrict__ h,
                                                      const int* __restrict__ src,
                                                      const int* __restrict__ dst,
                                                      const float* __restrict__ w,
                                                      const float* __restrict__ dinv,
                                                      float* __restrict__ out, int e) {
    unsigned gid = blockIdx.x * 256u + threadIdx.x;
    int ei = (int)(gid >> 4);
    if (ei >= e) return;
    int f4 = (int)(gid & 15u) * 4;
    int s = src[ei], d = dst[ei];
    float norm = dinv[s] * w[ei] * dinv[d];
    const float4 hv = *(const float4*)(h + (size_t)s * DFEAT + f4);
    float* o = out + (size_t)d * DFEAT + f4;
    atomicAdd(o + 0, hv.x * norm);
    atomicAdd(o + 1, hv.y * norm);
    atomicAdd(o + 2, hv.z * norm);
    atomicAdd(o + 3, hv.w * norm);
}

// ---------------- batchnorm ----------------
__global__ __launch_bounds__(128) void k_zero_stats(float* __restrict__ stats) {
    stats[threadIdx.x] = 0.0f;
}

__global__ __launch_bounds__(256) void k_bn_stats(const float* __restrict__ agg,
                                                  const float* __restrict__ b1,
                                                  float* __restrict__ stats, int n) {
    __shared__ float sh[512];
    const int f = threadIdx.x & 63;
    const int g = threadIdx.x >> 6;   // 0..3 row groups
    float s = 0.0f, ss = 0.0f;
    const float bias = b1[f];
    for (int i = blockIdx.x * 4 + g; i < n; i += gridDim.x * 4) {
        float v = agg[(size_t)i * DFEAT + f] + bias;
        s += v; ss += v * v;
    }
    sh[threadIdx.x] = s;
    sh[256 + threadIdx.x] = ss;
    __syncthreads();
    if (threadIdx.x < 64) {
        s  = sh[f]       + sh[64+f]  + sh[128+f] + sh[192+f];
        ss = sh[256+f]   + sh[320+f] + sh[384+f] + sh[448+f];
        atomicAdd(&stats[f], s);
        atomicAdd(&stats[64 + f], ss);
    }
}

__global__ __launch_bounds__(64) void k_bn_final(float* __restrict__ stats,
                                                 const float* __restrict__ gamma,
                                                 const float* __restrict__ beta,
                                                 float inv_n) {
    int f = threadIdx.x;
    float mu  = stats[f] * inv_n;
    float var = stats[64 + f] * inv_n - mu * mu;
    float rstd = rsqrtf(var + 1e-5f);
    float sc = gamma[f] * rstd;
    stats[f]      = sc;                 // scale
    stats[64 + f] = beta[f] - mu * sc;  // shift
}

__global__ __launch_bounds__(256) void k_norm_relu(const float* __restrict__ agg,
                                                   const float* __restrict__ b1,
                                                   const float* __restrict__ stats,
                                                   float* __restrict__ out, int total) {
    int i = blockIdx.x * 256 + threadIdx.x;
    if (i >= total) return;
    int f = i & 63;
    float v = (agg[i] + b1[f]) * stats[f] + stats[64 + f];
    out[i] = v > 0.0f ? v : 0.0f;
}

static inline int cdiv(int a, int b) { return (a + b - 1) / b; }

extern "C" void kernel_launch(void* const* d_in, const int* in_sizes, int n_in,
                              void* d_out, int out_size, void* d_ws, size_t ws_size,
                              hipStream_t stream) {
    const float* x     = (const float*)d_in[0];
    const int*   ei    = (const int*)  d_in[1];
    const float* ef    = (const float*)d_in[2];
    const float* W1    = (const float*)d_in[3];
    const float* b1    = (const float*)d_in[4];
    const float* gamma = (const float*)d_in[5];
    const float* beta  = (const float*)d_in[6];
    const float* W2    = (const float*)d_in[7];
    const float* b2    = (const float*)d_in[8];

    const int N = in_sizes[0] / DFEAT;
    const int E = in_sizes[1] / 2;
    const int* src = ei;
    const int* dst = ei + E;

    float* out   = (float*)d_out;
    float* dinv  = (float*)d_ws;          // N floats (deg -> dinv in place)
    float* stats = dinv + N;              // 128 floats
    float* bufB  = stats + 128;           // N*64 floats (agg1 / h2)
    float* bufA  = out;                   // N*64 floats (h1 / normalized h1)

    // degree + rsqrt
    k_init_deg<<<cdiv(N, 256), 256, 0, stream>>>(dinv, N);
    k_edge_deg<<<cdiv(E, 256), 256, 0, stream>>>(dst, ef, dinv, E);
    k_dinv    <<<cdiv(N, 256), 256, 0, stream>>>(dinv, N);

    // layer 1: h1 = X @ W1
    k_gemm64<<<cdiv(N, 64), 128, 0, stream>>>(x, W1, bufA, N);
    // aggregate: agg1 = scatter(h1 * norm) + selfloop
    k_selfloop    <<<cdiv(N * 16, 256), 256, 0, stream>>>(bufA, dinv, nullptr, bufB, N);
    k_edge_scatter<<<cdiv(E * 16, 256), 256, 0, stream>>>(bufA, src, dst, ef, dinv, bufB, E);

    // batchnorm stats over (agg1 + b1)
    k_zero_stats<<<1, 128, 0, stream>>>(stats);
    k_bn_stats  <<<512, 256, 0, stream>>>(bufB, b1, stats, N);
    k_bn_final  <<<1, 64, 0, stream>>>(stats, gamma, beta, 1.0f / (float)N);
    // normalized + relu -> bufA (h1 dead)
    k_norm_relu <<<cdiv(N * DFEAT, 256), 256, 0, stream>>>(bufB, b1, stats, bufA, N * DFEAT);

    // layer 2: h2 = h1n @ W2  (bufB dead -> reuse)
    k_gemm64<<<cdiv(N, 64), 128, 0, stream>>>(bufA, W2, bufB, N);
    // final aggregate into d_out (bias b2 added at init; scatter adds on top)
    k_selfloop    <<<cdiv(N * 16, 256), 256, 0, stream>>>(bufB, dinv, b2, out, N);
    k_edge_scatter<<<cdiv(E * 16, 256), 256, 0, stream>>>(bufB, src, dst, ef, dinv, out, E);
}